// ChamferDistance_48498770706994
// MI455X (gfx1250) — compile-verified
//
#include <hip/hip_runtime.h>

// CDNA5 / gfx1250 Chamfer distance.
// Core: V_WMMA_F32_16X16X4_F32 computes 16x16 tiles of (|q|^2 - 2 p.q) with
//   A row = (p.x, p.y, p.z, 1), B col = (-2q.x, -2q.y, -2q.z, |q|^2), K=4.
// |p|^2 is added after the row-min (monotone shift, argmin-invariant), as is
// the final max(d,0) clamp.
//
// WMMA f32 16x16x4 operand layout (ISA 7.12.2):
//   A (16x4): lane l, row = l&15; VGPR0/1 hold K=(0,1) for lanes 0-15,
//             K=(2,3) for lanes 16-31.
//   B (4x16): mirrored — lane l, col = l&15; VGPR0/1 hold K=(0,1)/(2,3)
//             split across lane halves.
//   C/D (16x16): VGPR v: rows v (lanes 0-15) / v+8 (lanes 16-31), col = l&15.

typedef __attribute__((ext_vector_type(2))) float v2f;
typedef __attribute__((ext_vector_type(8))) float v8f;

#define ROWTILES 4   // row tiles (of 16) per wave: A reuse per B load

// Pack (x,y,z) -> (x,y,z, x^2+y^2+z^2) as float4 for vector loads.
__global__ void chamfer_pack_kernel(const float* __restrict__ xyz,
                                    float4* __restrict__ packed, int npts) {
  int i = blockIdx.x * blockDim.x + threadIdx.x;
  if (i >= npts) return;
  float x = xyz[3 * i + 0];
  float y = xyz[3 * i + 1];
  float z = xyz[3 * i + 2];
  packed[i] = make_float4(x, y, z, x * x + y * y + z * z);
}

// For each of ROWTILES*16 rows (points of rowsP), find min/argmin of squared
// distance against all Mc points of colsP. One wave per ROWTILES row-tiles.
__global__ __launch_bounds__(256)
void chamfer_nn_kernel(const float* __restrict__ rowsP,   // [B][Nr][4]
                       const float* __restrict__ colsP,   // [B][Mc][4]
                       float* __restrict__ outDist,       // [B][Nr]
                       float* __restrict__ outIdx,        // [B][Nr] (values)
                       int Nr, int Mc) {
  const int lane = threadIdx.x & 31;
  const int wave = threadIdx.x >> 5;
  const int wavesPerBlock = blockDim.x >> 5;
  const int wid = blockIdx.x * wavesPerBlock + wave;   // global wave id

  const int grpsPerBatch = Nr / (16 * ROWTILES);
  const int b = wid / grpsPerBatch;
  const int rgrp = wid % grpsPerBatch;
  const int rowBase = rgrp * (16 * ROWTILES);

  const int rl = lane & 15;
  const bool hi = lane >= 16;          // lanes 16-31 carry K=2,3
  const float sgnB1 = hi ? 1.0f : -2.0f;

  // A operands: row point components (x,y | z,1) per lane half.
  v2f A[ROWTILES];
#pragma unroll
  for (int j = 0; j < ROWTILES; ++j) {
    float4 rp = ((const float4*)rowsP)[(size_t)b * Nr + rowBase + j * 16 + rl];
    A[j].x = hi ? rp.z : rp.x;
    A[j].y = hi ? 1.0f : rp.y;
  }

  // B source: float2 half of the packed float4 per column point.
  const float2* cbase = (const float2*)colsP + (size_t)b * Mc * 2 + (hi ? 1 : 0);

  float bestv[ROWTILES][8];
  int   besti[ROWTILES][8];
#pragma unroll
  for (int j = 0; j < ROWTILES; ++j)
#pragma unroll
    for (int i = 0; i < 8; ++i) { bestv[j][i] = 3.4e38f; besti[j][i] = 0; }

  // Software pipeline: keep one B tile in flight so the load of tile ct+16
  // overlaps the 4 WMMAs + min/argmin VALU of tile ct.
  float2 bl = cbase[(size_t)rl * 2];   // tile 0
  for (int ct = 0; ct < Mc; ct += 16) {
    int pf = ct + 16;
    pf = (pf < Mc) ? pf : 0;           // clamp: benign re-read of tile 0 last iter
    float2 nxt = cbase[(size_t)(pf + rl) * 2];

    v2f Bv;
    Bv.x = bl.x * -2.0f;     // -2x (lanes<16) or -2z (lanes>=16)
    Bv.y = bl.y * sgnB1;     // -2y (lanes<16) or |q|^2 (lanes>=16)
    const int m = ct + rl;   // column index carried by this lane

#pragma unroll
    for (int j = 0; j < ROWTILES; ++j) {
      v8f Cz = {};
      v8f d = __builtin_amdgcn_wmma_f32_16x16x4_f32(
          false, A[j], false, Bv, (short)0, Cz, false, false);
#pragma unroll
      for (int i = 0; i < 8; ++i) {
        float dv = d[i];
        bool lt = dv < bestv[j][i];            // strict <: keep first index
        bestv[j][i] = lt ? dv : bestv[j][i];
        besti[j][i] = lt ? m : besti[j][i];
      }
    }
    bl = nxt;
  }

  // Reduce across the 16 lanes of each half (columns of one row), keeping
  // the lowest index on value ties (matches jnp.argmin).
#pragma unroll
  for (int j = 0; j < ROWTILES; ++j) {
#pragma unroll
    for (int i = 0; i < 8; ++i) {
      float v = bestv[j][i];
      int ix = besti[j][i];
#pragma unroll
      for (int mask = 1; mask < 16; mask <<= 1) {
        float ov = __shfl_xor(v, mask, 32);
        int oi = __shfl_xor(ix, mask, 32);
        if (ov < v || (ov == v && oi < ix)) { v = ov; ix = oi; }
      }
      if (rl == 0) {  // lane 0 -> row i, lane 16 -> row i+8
        int row = rowBase + j * 16 + i + (hi ? 8 : 0);
        float n1 = rowsP[((size_t)b * Nr + row) * 4 + 3];
        float dist = v + n1;
        dist = dist > 0.0f ? dist : 0.0f;
        outDist[(size_t)b * Nr + row] = dist;
        outIdx[(size_t)b * Nr + row] = (float)ix;
      }
    }
  }
}

extern "C" void kernel_launch(void* const* d_in, const int* in_sizes, int n_in,
                              void* d_out, int out_size, void* d_ws, size_t ws_size,
                              hipStream_t stream) {
  const float* xyz1 = (const float*)d_in[0];
  const float* xyz2 = (const float*)d_in[1];
  const int B = 8;
  const int N = in_sizes[0] / (B * 3);   // 8192
  const int M = in_sizes[1] / (B * 3);   // 8192

  // Workspace: packed (x,y,z,|p|^2) float4 arrays, 1 MB each.
  float* p1 = (float*)d_ws;                 // B*N*4 floats
  float* p2 = p1 + (size_t)B * N * 4;       // B*M*4 floats

  // Output layout (flat, return order): dist1, dist2, idx1, idx2.
  float* dist1 = (float*)d_out;
  float* dist2 = dist1 + (size_t)B * N;
  float* idx1 = dist2 + (size_t)B * M;
  float* idx2 = idx1 + (size_t)B * N;

  chamfer_pack_kernel<<<(B * N + 255) / 256, 256, 0, stream>>>(
      xyz1, (float4*)p1, B * N);
  chamfer_pack_kernel<<<(B * M + 255) / 256, 256, 0, stream>>>(
      xyz2, (float4*)p2, B * M);

  // Pass 1: rows = set1, cols = set2  -> dist1/idx1
  {
    int waves = B * (N / (16 * ROWTILES));
    chamfer_nn_kernel<<<waves / 8, 256, 0, stream>>>(p1, p2, dist1, idx1, N, M);
  }
  // Pass 2: rows = set2, cols = set1  -> dist2/idx2
  {
    int waves = B * (M / (16 * ROWTILES));
    chamfer_nn_kernel<<<waves / 8, 256, 0, stream>>>(p2, p1, dist2, idx2, M, N);
  }
}